// OrthogonalProductQuantizer_89601607729712
// MI455X (gfx1250) — compile-verified
//
#include <hip/hip_runtime.h>

typedef __attribute__((ext_vector_type(2))) float v2f;
typedef __attribute__((ext_vector_type(8))) float v8f;

#define NHEADS   8
#define KCODES   1024
#define HD       128
#define BATCHSZ  32768
#define MB       128            // rows per block
#define NCH      64             // codes per LDS chunk
#define BSTR     132            // LDS row stride in floats (132 % 64 == 4 -> conflict-free frags)
#define NCHUNKS  (KCODES / NCH) // 16
#define KQ       (HD / 4)       // 32 k-quads per 16x16 tile

#define IDX_OFF  ((size_t)BATCHSZ * 1024)               // z_q is 32768*1024 floats
#define DIST_OFF (IDX_OFF + (size_t)BATCHSZ * NHEADS)   // indices are 32768*8 ints

__global__ __launch_bounds__(256)
void pq_fused_kernel(const float* __restrict__ z,
                     const float* __restrict__ cb,
                     float* __restrict__ out)
{
    __shared__ float Blds[NCH * BSTR];
    __shared__ float csq[NCH];
    __shared__ int   idx_lds[MB];

    const int t    = threadIdx.x;
    const int lane = t & 31;
    const int w    = t >> 5;        // wave id 0..7 == M-tile id
    const int h    = blockIdx.y;
    const int m0   = blockIdx.x * MB;
    const int lh   = lane >> 4;     // lane half (0/1)
    const int l15  = lane & 15;

    // ---- preload this wave's 16x128 A tile as 32 WMMA f32 16x16x4 A fragments ----
    // A layout (ISA): lanes 0-15 hold M=lane, VGPR0/1 = K=k0,k0+1; lanes 16-31: K=k0+2,k0+3
    v2f afrag[KQ];
    {
        const float* zp = z + (size_t)(m0 + w * 16 + l15) * (NHEADS * HD) + h * HD + lh * 2;
#pragma unroll
        for (int k = 0; k < KQ; ++k)
            afrag[k] = *(const v2f*)(zp + k * 4);
    }

    // ---- ||z||^2 per row: each lane holds half the row; combine across lane halves ----
    float zsq;
    {
        float p = 0.f;
#pragma unroll
        for (int k = 0; k < KQ; ++k)
            p += afrag[k].x * afrag[k].x + afrag[k].y * afrag[k].y;
        zsq = p + __shfl_xor(p, 16, 32);
    }
    // zsq_r[r]: the ||z||^2 for the row this lane touches in accumulator VGPR r
    float zsq_r[8];
#pragma unroll
    for (int r = 0; r < 8; ++r)
        zsq_r[r] = __shfl(zsq, r + (lh << 3), 32);

    float minv[8];
    int   mini[8];
#pragma unroll
    for (int r = 0; r < 8; ++r) { minv[r] = __builtin_inff(); mini[r] = 0; }

    float* dist_out = out + DIST_OFF;

    for (int c = 0; c < NCHUNKS; ++c) {
        const int cbase = c * NCH;
        __syncthreads();
        // cooperative load of 64 codebook rows (codes cbase..cbase+63) into LDS
        {
            const float* cbp = cb + ((size_t)h * KCODES + cbase) * HD;
#pragma unroll
            for (int j = 0; j < 8; ++j) {
                int i = t + 256 * j;
                int n = i >> 5, d = (i & 31) * 4;
                *(float4*)&Blds[n * BSTR + d] = *(const float4*)(cbp + (size_t)n * HD + d);
            }
        }
        __syncthreads();
        // ||c||^2 for this chunk
        if (t < NCH) {
            float s = 0.f;
            const float* bp = &Blds[t * BSTR];
#pragma unroll
            for (int d = 0; d < HD; d += 4) {
                float4 v = *(const float4*)(bp + d);
                s += v.x * v.x + v.y * v.y + v.z * v.z + v.w * v.w;
            }
            csq[t] = s;
        }
        __syncthreads();

        // 4 N-tiles of 16 per chunk; this wave covers all of them for its M-tile
        for (int tn = 0; tn < NCH / 16; ++tn) {
            const int nloc = tn * 16 + l15;
            const float cs = csq[nloc];
            // B layout (4x16): lanes 0-15: K=k0,k0+1 for N=lane; lanes 16-31: K=k0+2,k0+3
            const float* bp = &Blds[nloc * BSTR + lh * 2];
            v8f acc = {};
#pragma unroll
            for (int k = 0; k < KQ; ++k) {
                v2f b = *(const v2f*)(bp + k * 4);
                acc = __builtin_amdgcn_wmma_f32_16x16x4_f32(
                          false, afrag[k], false, b, (short)0, acc, false, false);
            }
            const int ncol = cbase + nloc;
#pragma unroll
            for (int r = 0; r < 8; ++r) {
                float d  = fmaf(-2.f, acc[r], cs) + zsq_r[r];
                int   gb = m0 + w * 16 + r + (lh << 3);
                __builtin_nontemporal_store(
                    d, &dist_out[((size_t)gb * NHEADS + h) * KCODES + ncol]);
                if (d < minv[r]) { minv[r] = d; mini[r] = ncol; }
            }
        }
    }

    // ---- argmin: butterfly reduce within each 16-lane half (tie -> lowest index) ----
#pragma unroll
    for (int r = 0; r < 8; ++r) {
        float mv = minv[r];
        int   mi = mini[r];
#pragma unroll
        for (int off = 8; off >= 1; off >>= 1) {
            float ov = __shfl_xor(mv, off, 16);
            int   oi = __shfl_xor(mi, off, 16);
            if (ov < mv || (ov == mv && oi < mi)) { mv = ov; mi = oi; }
        }
        if (l15 == 0) {
            int rowl = w * 16 + r + (lh << 3);
            int gb   = m0 + rowl;
            ((int*)out)[IDX_OFF + (size_t)gb * NHEADS + h] = mi;
            idx_lds[rowl] = mi;
        }
    }
    __syncthreads();

    // ---- gather z_q rows from codebook (L2-resident) ----
#pragma unroll
    for (int j = 0; j < 16; ++j) {
        int i   = t + 256 * j;
        int row = i >> 5, d = (i & 31) * 4;
        int code = idx_lds[row];
        float4 v = *(const float4*)(cb + ((size_t)h * KCODES + code) * HD + d);
        *(float4*)&out[(size_t)(m0 + row) * (NHEADS * HD) + h * HD + d] = v;
    }
}

extern "C" void kernel_launch(void* const* d_in, const int* in_sizes, int n_in,
                              void* d_out, int out_size, void* d_ws, size_t ws_size,
                              hipStream_t stream)
{
    const float* z  = (const float*)d_in[0];
    const float* cb = (const float*)d_in[1];
    float* out = (float*)d_out;
    dim3 grid(BATCHSZ / MB, NHEADS);
    pq_fused_kernel<<<grid, 256, 0, stream>>>(z, cb, out);
}